// VQVAE_62629213110907
// MI455X (gfx1250) — compile-verified
//
#include <hip/hip_runtime.h>

typedef float v2f __attribute__((ext_vector_type(2)));
typedef float v8f __attribute__((ext_vector_type(8)));
typedef unsigned int u32x4 __attribute__((ext_vector_type(4)));
typedef int i32x4 __attribute__((ext_vector_type(4)));
typedef int i32x8 __attribute__((ext_vector_type(8)));

#define KCODES 512
#define EMB 64
#define LDS_STRIDE 68          // padded row (floats): conflict-free banks, 16B-aligned rows
#define WAVES_PER_BLOCK 16
#define THREADS (WAVES_PER_BLOCK * 32)
#define NVEC 131072            // 32*64*64
#define NPART (2 * (NVEC / 16))  // one partial per (wave, half-wave) = 16384

__global__ __launch_bounds__(THREADS)
void vq_main(const float* __restrict__ x,
             const float* __restrict__ emb,
             float* __restrict__ out,        // 8388608 floats, (B,D,H,W)
             float* __restrict__ idx_out,    // 131072 floats (indices as float)
             float* __restrict__ parts)      // NPART loss partials in ws
{
    extern __shared__ float lds[];
    float* lds_emb = lds;                                 // KCODES*LDS_STRIDE (offset 0)
    float* lds_esq = lds + KCODES * LDS_STRIDE;           // KCODES
    int*   lds_idx = (int*)(lds_esq + KCODES);            // WAVES_PER_BLOCK*16
    float* lds_xsq = (float*)(lds_idx + WAVES_PER_BLOCK * 16); // WAVES_PER_BLOCK*16

    const int tid = threadIdx.x;

#if __has_builtin(__builtin_amdgcn_tensor_load_to_lds)
    // ---- TDM: DMA the 512x64 f32 codebook into LDS with hardware row padding ----
    // pad_interval code 5 -> pad after every 64 DWORDs (one 256B row)
    // pad_amount  code 3 -> insert 4 DWORDs  => LDS row stride = 272B = LDS_STRIDE
    if (tid < 32) {
        const unsigned long long ga = (unsigned long long)emb;
        u32x4 g0;
        g0[0] = 1u;                                        // count=1, user mode, no gather
        g0[1] = 0u;                                        // lds_addr = 0 (lds_emb)
        g0[2] = (unsigned)(ga & 0xFFFFFFFFu);              // global_addr[31:0]
        g0[3] = (unsigned)((ga >> 32) & 0x01FFFFFFu)       // global_addr[56:32]
              | (2u << 30);                                // type = 2 ("image")
        i32x8 g1;
        g1[0] = (2 << 16)      // data_size = 4 bytes
              | (1 << 20)      // pad_enable
              | (5 << 22)      // pad_interval: 64 DWORDs
              | (3 << 25);     // pad_amount: 4 DWORDs
        g1[1] = (int)(EMB << 16);      // tensor_dim0 = 64 (bits 79:48, low half)
        g1[2] = (int)(KCODES << 16);   // tensor_dim0 hi=0 | tensor_dim1 = 512 (low half)
        g1[3] = (int)(EMB << 16);      // tensor_dim1 hi=0 | tile_dim0 = 64
        g1[4] = KCODES;                // tile_dim1 = 512 | tile_dim2 = 0
        g1[5] = EMB;                   // tensor_dim0_stride[31:0] = 64 elements
        g1[6] = 0;                     // stride hi | tensor_dim1_stride lo
        g1[7] = 0;
        const i32x4 z4 = {0, 0, 0, 0};                     // groups 2/3 unused (2D)
        const i32x8 z8 = {0, 0, 0, 0, 0, 0, 0, 0};
        __builtin_amdgcn_tensor_load_to_lds(g0, g1, z4, z4, z8, 0);
        __builtin_amdgcn_s_wait_tensorcnt(0);
    }
#else
    // ---- fallback: cooperative load codebook -> LDS (padded rows) ----
    for (int i = tid; i < KCODES * (EMB / 4); i += THREADS) {
        const int row = i >> 4;
        const int c4  = i & 15;
        const float4 v = ((const float4*)emb)[i];
        float4* dst = (float4*)(lds_emb + row * LDS_STRIDE + c4 * 4);
        *dst = v;
    }
#endif
    __syncthreads();
    if (tid < KCODES) {
        float s = 0.f;
        const float* rp = lds_emb + tid * LDS_STRIDE;
        #pragma unroll
        for (int d = 0; d < EMB; ++d) s += rp[d] * rp[d];
        lds_esq[tid] = s;
    }
    __syncthreads();

    const int lane = tid & 31;
    const int wave = tid >> 5;
    const int hl   = lane >> 4;     // half-wave select (0/1)
    const int r    = lane & 15;     // row within A-tile == column within C-tile

    const int gwave = blockIdx.x * WAVES_PER_BLOCK + wave;
    const int n0 = gwave * 16;              // 16 consecutive vectors, same (b,h)
    const int w0 = n0 & 63;
    const int h  = (n0 >> 6) & 63;
    const int b  = n0 >> 12;
    const long rowbase = (long)b * 262144 + (long)h * 64 + w0;  // + d*4096 + r

    // ---- load A: 16 rows x 64 K as 16 chunks of K=4 (WMMA f32 A layout) ----
    v2f a[16];
    float xsq = 0.f;   // this lane's partial sum-of-squares for row r
    #pragma unroll
    for (int kc = 0; kc < 16; ++kc) {
        const int d0 = 4 * kc + (hl ? 2 : 0);
        const int d1 = 4 * kc + (hl ? 3 : 1);
        const float a0 = x[rowbase + (long)d0 * 4096 + r];
        const float a1 = x[rowbase + (long)d1 * 4096 + r];
        a[kc][0] = a0; a[kc][1] = a1;
        xsq += a0 * a0 + a1 * a1;
    }

    float bestv[8];
    int   besti[8];
    #pragma unroll
    for (int j = 0; j < 8; ++j) { bestv[j] = 3.4e38f; besti[j] = 0; }

    // ---- sweep 32 code tiles of 16 codes each ----
    for (int c0 = 0; c0 < KCODES; c0 += 16) {
        v8f acc = {0.f, 0.f, 0.f, 0.f, 0.f, 0.f, 0.f, 0.f};
        const float* brow = lds_emb + (c0 + r) * LDS_STRIDE;  // this lane's code row
        #pragma unroll
        for (int kc = 0; kc < 16; ++kc) {
            const float2 bv = *(const float2*)(brow + 4 * kc + 2 * hl);
            v2f bb; bb[0] = bv.x; bb[1] = bv.y;
            acc = __builtin_amdgcn_wmma_f32_16x16x4_f32(
                false, a[kc], false, bb, (short)0, acc, false, false);
        }
        const float esq  = lds_esq[c0 + r];   // same addr both halves -> LDS broadcast
        const int   code = c0 + r;
        #pragma unroll
        for (int j = 0; j < 8; ++j) {
            const float dist = esq - 2.0f * acc[j];   // argmin-equivalent distance
            if (dist < bestv[j]) { bestv[j] = dist; besti[j] = code; }
        }
    }

    // ---- cross-lane argmin within each 16-lane half (rows j / j+8) ----
    #pragma unroll
    for (int j = 0; j < 8; ++j) {
        #pragma unroll
        for (int off = 8; off > 0; off >>= 1) {
            const float ov = __shfl_xor(bestv[j], off, 32);
            const int   oi = __shfl_xor(besti[j], off, 32);
            if (ov < bestv[j] || (ov == bestv[j] && oi < besti[j])) {
                bestv[j] = ov; besti[j] = oi;
            }
        }
    }

    // full ||x||^2 per row: halves at lanes r and r+16
    const float xsqf = xsq + __shfl_xor(xsq, 16, 32);
    lds_xsq[wave * 16 + r] = xsqf;      // both halves write identical value

    // lane 0 holds rows 0..7, lane 16 holds rows 8..15
    if (r == 0) {
        const int rbase = hl * 8;
        float lpart = 0.f;
        #pragma unroll
        for (int j = 0; j < 8; ++j) {
            const int rowm = rbase + j;
            lds_idx[wave * 16 + rowm] = besti[j];
            idx_out[n0 + rowm] = (float)besti[j];
            // ||x - e*||^2 = ||x||^2 + (e_sq* - 2 dot*)
            lpart += lds_xsq[wave * 16 + rowm] + bestv[j];
        }
        parts[gwave * 2 + hl] = lpart;  // deterministic (no atomics)
    }

    // ---- gather x_q from LDS codebook, write to (B,D,H,W) ----
    const int myidx = lds_idx[wave * 16 + r];   // same-wave LDS RAW: in-order
    const float* crow = lds_emb + myidx * LDS_STRIDE;
    #pragma unroll
    for (int d2 = 0; d2 < 32; ++d2) {
        const int d = 2 * d2 + hl;              // 64B-coalesced per half-wave
        out[rowbase + (long)d * 4096 + r] = crow[d];
    }
}

__global__ __launch_bounds__(256)
void vq_finalize(const float* __restrict__ parts, float* __restrict__ loss_out)
{
    __shared__ float sbuf[256];
    float s = 0.f;
    for (int i = threadIdx.x; i < NPART; i += 256) s += parts[i];
    sbuf[threadIdx.x] = s;
    __syncthreads();
    for (int off = 128; off > 0; off >>= 1) {
        if ((int)threadIdx.x < off) sbuf[threadIdx.x] += sbuf[threadIdx.x + off];
        __syncthreads();
    }
    if (threadIdx.x == 0)
        loss_out[0] = 1.5f * sbuf[0] / (131072.0f * 64.0f);  // (1+BETA)*MSE
}

extern "C" void kernel_launch(void* const* d_in, const int* in_sizes, int n_in,
                              void* d_out, int out_size, void* d_ws, size_t ws_size,
                              hipStream_t stream) {
    const float* x   = (const float*)d_in[0];   // (32,64,64,64) f32
    const float* emb = (const float*)d_in[1];   // (512,64) f32
    float* out  = (float*)d_out;                // [0, 8388608): out
    float* loss = out + 8388608;                // [8388608]: loss
    float* idxo = out + 8388609;                // [8388609, ...): indices (as float)
    float* parts = (float*)d_ws;                // NPART loss partials

    const size_t shmem = (size_t)(KCODES * LDS_STRIDE + KCODES) * 4
                       + (size_t)WAVES_PER_BLOCK * 16 * 4 * 2;   // ~143 KB

    const int blocks = NVEC / (WAVES_PER_BLOCK * 16);            // 512
    vq_main<<<blocks, THREADS, shmem, stream>>>(x, emb, out, idxo, parts);
    vq_finalize<<<1, 256, 0, stream>>>(parts, loss);
}